// TorchWrithe_66056597012864
// MI455X (gfx1250) — compile-verified
//
#include <hip/hip_runtime.h>
#include <hip/hip_bf16.h>
#include <math.h>

// ---------------------------------------------------------------------------
// Writhe kernel for MI455X (gfx1250).
// Structure: one block = (one frame, 256 segments). The frame's 512x3 f32
// coordinates (6144 B) are DMA'd into LDS with a single Tensor Data Mover
// descriptor (tensor_load_to_lds), then each thread computes one writhe value.
// Output-bandwidth bound (~33 MB stores), so we keep loads LDS-resident and
// stores/segment-loads fully coalesced.
// ---------------------------------------------------------------------------

#define ATOMS        512
#define FRAME_FLOATS (ATOMS * 3)          // 1536 dwords = 6144 bytes per frame
#define BLOCK        256

typedef __attribute__((ext_vector_type(4))) unsigned int tdm_g0_t;  // D# group0: 4 SGPRs
typedef __attribute__((ext_vector_type(8))) unsigned int tdm_g1_t;  // D# group1: 8 SGPRs

struct F3 { float x, y, z; };

__device__ __forceinline__ F3 f3sub(F3 a, F3 b) { return {a.x - b.x, a.y - b.y, a.z - b.z}; }
__device__ __forceinline__ float f3dot(F3 a, F3 b) { return a.x * b.x + a.y * b.y + a.z * b.z; }
__device__ __forceinline__ F3 f3cross(F3 a, F3 b) {
    return {a.y * b.z - a.z * b.y, a.z * b.x - a.x * b.z, a.x * b.y - a.y * b.x};
}
__device__ __forceinline__ F3 f3unit(F3 a) {
    float r = rsqrtf(f3dot(a, a));
    return {a.x * r, a.y * r, a.z * r};
}
__device__ __forceinline__ float asin_dot(F3 a, F3 b) {
    float d = f3dot(a, b);
    d = fminf(1.0f, fmaxf(-1.0f, d));
    return asinf(d);
}

__global__ __launch_bounds__(BLOCK) void writhe_kernel(
    const float* __restrict__ xyz,   // [nframes, 512, 3] f32
    const int*   __restrict__ seg,   // [nseg, 4] i32
    float*       __restrict__ out,   // [nframes, nseg] f32
    int nseg)
{
    __shared__ float P[FRAME_FLOATS];

    const int frame = blockIdx.y;

    // ---- TDM: async-load this frame's 6144 B into LDS (wave 0 issues) ----
    if (threadIdx.x < 32) {   // wave-uniform guard -> scalar branch; TDM ignores EXEC
        unsigned long long gaddr =
            (unsigned long long)(const void*)(xyz + (size_t)frame * FRAME_FLOATS);
        unsigned lds_base = (unsigned)(unsigned long long)(void*)&P[0]; // low 32b = LDS offset

        tdm_g0_t g0;
        g0.x = 1u;                                          // count=1, user descriptor
        g0.y = lds_base;                                    // lds_addr (bytes)
        g0.z = (unsigned)(gaddr & 0xFFFFFFFFull);           // global_addr[31:0]
        g0.w = (unsigned)((gaddr >> 32) & 0x01FFFFFFull)    // global_addr[56:32]
             | (2u << 30);                                  // type = 2 ("image")

        tdm_g1_t g1;
        g1.s0 = (2u << 16);                                 // data_size = 4 bytes; wg_mask=0
        g1.s1 = ((unsigned)FRAME_FLOATS & 0xFFFFu) << 16;   // tensor_dim0[15:0] @ bits[63:48]
        g1.s2 = (((unsigned)FRAME_FLOATS >> 16) & 0xFFFFu)  // tensor_dim0[31:16]
              | (1u << 16);                                 // tensor_dim1 = 1
        g1.s3 = ((unsigned)FRAME_FLOATS & 0xFFFFu) << 16;   // tile_dim0 = 1536 @ bits[127:112]
        g1.s4 = 1u;                                         // tile_dim1 = 1, tile_dim2 = 0
        g1.s5 = (unsigned)FRAME_FLOATS;                     // tensor_dim0_stride[31:0]
        g1.s6 = 0u;                                         // stride hi / dim1_stride lo
        g1.s7 = 0u;

        // 2-SGPR-group form (tensor up to 2D); VADDR2/VADDR3 = NULL.
        asm volatile("tensor_load_to_lds %0, %1" :: "s"(g0), "s"(g1) : "memory");
        __builtin_amdgcn_s_wait_tensorcnt(0);
    }
    __syncthreads();   // make TDM-written LDS visible to all 8 waves

    const int s = blockIdx.x * BLOCK + threadIdx.x;
    if (s >= nseg) return;

    // Coalesced 16B load of the 4 atom indices for this segment.
    const int4 q = ((const int4*)seg)[s];

    F3 p1 = {P[3 * q.x], P[3 * q.x + 1], P[3 * q.x + 2]};
    F3 p2 = {P[3 * q.y], P[3 * q.y + 1], P[3 * q.y + 2]};
    F3 p3 = {P[3 * q.z], P[3 * q.z + 1], P[3 * q.z + 2]};
    F3 p4 = {P[3 * q.w], P[3 * q.w + 1], P[3 * q.w + 2]};

    F3 r12 = f3sub(p2, p1);
    F3 r13 = f3sub(p3, p1);
    F3 r14 = f3sub(p4, p1);
    F3 r23 = f3sub(p3, p2);
    F3 r24 = f3sub(p4, p2);
    F3 r34 = f3sub(p4, p3);

    F3 n1 = f3unit(f3cross(r13, r14));
    F3 n2 = f3unit(f3cross(r14, r24));
    F3 n3 = f3unit(f3cross(r24, r23));
    F3 n4 = f3unit(f3cross(r23, r13));

    float theta = asin_dot(n1, n2) + asin_dot(n2, n3)
                + asin_dot(n3, n4) + asin_dot(n4, n1);

    float t  = f3dot(f3cross(r34, r12), r13);
    float sg = (t > 0.0f) ? 1.0f : ((t < 0.0f) ? -1.0f : 0.0f);

    const float inv2pi = 0.15915494309189535f;  // 1/(2*pi)
    out[(size_t)frame * nseg + s] = theta * sg * inv2pi;
}

extern "C" void kernel_launch(void* const* d_in, const int* in_sizes, int n_in,
                              void* d_out, int out_size, void* d_ws, size_t ws_size,
                              hipStream_t stream) {
    const float* xyz = (const float*)d_in[0];          // [nframes, 512, 3] f32
    const int*   seg = (const int*)d_in[1];            // [nseg, 4] i32
    float*       out = (float*)d_out;                  // [nframes, nseg] f32

    const int nseg    = in_sizes[1] / 4;
    const int nframes = in_sizes[0] / FRAME_FLOATS;

    dim3 block(BLOCK);
    dim3 grid((nseg + BLOCK - 1) / BLOCK, nframes);
    writhe_kernel<<<grid, block, 0, stream>>>(xyz, seg, out, nseg);
}